// StructureLearning_9259949490946
// MI455X (gfx1250) — compile-verified
//
#include <hip/hip_runtime.h>
#include <hip/hip_bf16.h>

typedef __attribute__((ext_vector_type(2))) float v2f;
typedef __attribute__((ext_vector_type(8))) float v8f;

#define DUSER   8192
#define DPOST   8192
#define NVEC    128
#define TILEM   256
#define TILEN   128
#define KCHUNK  32
#define LPITCH  36   // dword pitch: %4==0 (16B LDS stores), conflict-free frag reads

// ---------------------------------------------------------------------------
// Phase 1: out[row,:] = normalize(attr[row,:] * W[row,:]) with eps floor.
// One wave32 per row, float4 per lane, shfl_xor reduction.
// ---------------------------------------------------------------------------
__global__ __launch_bounds__(256) void sl_normalize_rows(
    const float* __restrict__ attr, const float* __restrict__ W,
    float* __restrict__ out, int rows)
{
    const int row  = blockIdx.x * 8 + (threadIdx.x >> 5);
    const int lane = threadIdx.x & 31;
    if (row >= rows) return;

    const float4* a4 = (const float4*)(attr + (size_t)row * NVEC);
    const float4* w4 = (const float4*)(W    + (size_t)row * NVEC);
    float4 a = a4[lane];
    float4 w = w4[lane];
    float4 v = make_float4(a.x * w.x, a.y * w.y, a.z * w.z, a.w * w.w);

    float ss = v.x * v.x + v.y * v.y + v.z * v.z + v.w * v.w;
#pragma unroll
    for (int off = 16; off >= 1; off >>= 1)
        ss += __shfl_xor(ss, off, 32);

    const float inv = 1.0f / fmaxf(sqrtf(ss), 1e-8f);
    float4 o = make_float4(v.x * inv, v.y * inv, v.z * inv, v.w * inv);
    ((float4*)(out + (size_t)row * NVEC))[lane] = o;
}

// ---------------------------------------------------------------------------
// Phase 2: C = threshold( X @ Y^T ), X:[8192,128], Y:[8192,128] row-major,
// via V_WMMA_F32_16X16X4_F32. Block = 256 threads (8 waves) computing a
// 256x128 output tile; each wave owns 64x64 (4x4 v8f accumulators).
// Per k-step: 8 ds_load_b64 feed 16 WMMAs (1:2 load:math ratio).
// ---------------------------------------------------------------------------
__global__ __launch_bounds__(256) void sl_cosgemm_thresh(
    const float* __restrict__ X, const float* __restrict__ Y,
    float* __restrict__ out, float thresh)
{
    __shared__ float Xs[TILEM * LPITCH];   // 36,864 B
    __shared__ float Ys[TILEN * LPITCH];   // 18,432 B

    const int m0   = blockIdx.y * TILEM;
    const int n0   = blockIdx.x * TILEN;
    const int tid  = threadIdx.x;
    const int lane = tid & 31;
    const int wave = tid >> 5;
    const int wm   = (wave >> 1) * 64;   // 0,64,128,192
    const int wn   = (wave & 1)  * 64;   // 0,64
    const int lrow = lane & 15;          // M (A frag) / N (B frag) within 16
    const int lk   = (lane >> 4) * 2;    // lanes 16-31 hold K+2,K+3

    v8f acc[4][4];
#pragma unroll
    for (int i = 0; i < 4; ++i)
#pragma unroll
        for (int j = 0; j < 4; ++j)
            acc[i][j] = (v8f){0.f, 0.f, 0.f, 0.f, 0.f, 0.f, 0.f, 0.f};

    for (int kc = 0; kc < NVEC; kc += KCHUNK) {
        // Stage X chunk: 256 rows x 32 cols = 2048 float4 (8 per thread).
#pragma unroll
        for (int it = 0; it < 8; ++it) {
            const int idx = tid + it * 256;
            const int r   = idx >> 3;        // 0..255
            const int c4  = (idx & 7) * 4;   // 0..28 dwords
            float4 xv = *(const float4*)(X + (size_t)(m0 + r) * NVEC + kc + c4);
            *(float4*)&Xs[r * LPITCH + c4] = xv;
        }
        // Stage Y chunk: 128 rows x 32 cols = 1024 float4 (4 per thread).
#pragma unroll
        for (int it = 0; it < 4; ++it) {
            const int idx = tid + it * 256;
            const int r   = idx >> 3;        // 0..127
            const int c4  = (idx & 7) * 4;
            float4 yv = *(const float4*)(Y + (size_t)(n0 + r) * NVEC + kc + c4);
            *(float4*)&Ys[r * LPITCH + c4] = yv;
        }
        // Prefetch next chunk's lines toward cache while we compute.
        if (kc + KCHUNK < NVEC) {
#pragma unroll
            for (int it = 0; it < 8; ++it) {
                const int idx = tid + it * 256;
                __builtin_prefetch(X + (size_t)(m0 + (idx >> 3)) * NVEC +
                                   kc + KCHUNK + (idx & 7) * 4, 0, 1);
            }
#pragma unroll
            for (int it = 0; it < 4; ++it) {
                const int idx = tid + it * 256;
                __builtin_prefetch(Y + (size_t)(n0 + (idx >> 3)) * NVEC +
                                   kc + KCHUNK + (idx & 7) * 4, 0, 1);
            }
        }
        __syncthreads();

#pragma unroll
        for (int kk = 0; kk < KCHUNK; kk += 4) {
            v2f afr[4], bfr[4];
#pragma unroll
            for (int i = 0; i < 4; ++i)
                afr[i] = *(const v2f*)&Xs[(wm + i * 16 + lrow) * LPITCH + kk + lk];
#pragma unroll
            for (int j = 0; j < 4; ++j)
                bfr[j] = *(const v2f*)&Ys[(wn + j * 16 + lrow) * LPITCH + kk + lk];

#pragma unroll
            for (int i = 0; i < 4; ++i)
#pragma unroll
                for (int j = 0; j < 4; ++j)
                    acc[i][j] = __builtin_amdgcn_wmma_f32_16x16x4_f32(
                        /*neg_a=*/false, afr[i], /*neg_b=*/false, bfr[j],
                        /*c_mod=*/(short)0, acc[i][j],
                        /*reuse_a=*/false, /*reuse_b=*/false);
        }
        __syncthreads();
    }

    // Epilogue: C/D layout -> M = r + 8*(lane>=16), N = lane&15. Threshold+store.
    const int mhalf = 8 * (lane >> 4);
    const int nbase = n0 + wn + (lane & 15);
#pragma unroll
    for (int i = 0; i < 4; ++i) {
#pragma unroll
        for (int j = 0; j < 4; ++j) {
#pragma unroll
            for (int r = 0; r < 8; ++r) {
                float v = acc[i][j][r];
                v = (v < thresh) ? 0.0f : v;
                const size_t m = (size_t)(m0 + wm + i * 16 + mhalf + r);
                out[m * (size_t)DPOST + (nbase + j * 16)] = v;
            }
        }
    }
}

// ---------------------------------------------------------------------------
// Launch: normalize A,B into d_ws, then sim_uu = A@A^T and sim_up = A@B^T.
// d_out = [sim_uu (8192*8192 f32), sim_up (8192*8192 f32)].
// ---------------------------------------------------------------------------
extern "C" void kernel_launch(void* const* d_in, const int* in_sizes, int n_in,
                              void* d_out, int out_size, void* d_ws, size_t ws_size,
                              hipStream_t stream) {
    const float* attr1 = (const float*)d_in[0];
    const float* attr2 = (const float*)d_in[1];
    const float* W1    = (const float*)d_in[2];
    const float* W2    = (const float*)d_in[3];
    float* out  = (float*)d_out;
    float* Ahat = (float*)d_ws;                       // 8192*128 f32
    float* Bhat = Ahat + (size_t)DUSER * NVEC;        // 8192*128 f32 (8 MB total)

    sl_normalize_rows<<<DUSER / 8, 256, 0, stream>>>(attr1, W1, Ahat, DUSER);
    sl_normalize_rows<<<DPOST / 8, 256, 0, stream>>>(attr2, W2, Bhat, DPOST);

    dim3 grid(DPOST / TILEN, DUSER / TILEM);   // (64, 32)
    // sim_uu = Ahat @ Ahat^T (first in output)
    sl_cosgemm_thresh<<<grid, 256, 0, stream>>>(Ahat, Ahat, out, 0.9f);
    // sim_up = Ahat @ Bhat^T (second)
    sl_cosgemm_thresh<<<grid, 256, 0, stream>>>(Ahat, Bhat,
                                                out + (size_t)DUSER * DPOST, 0.9f);
}